// PolaLinearAttention_65180423685629
// MI455X (gfx1250) — compile-verified
//
#include <hip/hip_runtime.h>
#include <hip/hip_bf16.h>
#include <math.h>

typedef float v2f __attribute__((ext_vector_type(2)));
typedef float v8f __attribute__((ext_vector_type(8)));

#define WMMA_F32(acc, a, b) \
  acc = __builtin_amdgcn_wmma_f32_16x16x4_f32(false, a, false, b, (short)0, acc, false, false)

constexpr int B_ = 8, N_ = 4096, C_ = 512;
constexpr int BN = B_ * N_;                  // 32768
constexpr float ALPHA = 4.0f;
constexpr float EPS_ = 1e-6f;

__device__ __forceinline__ float softplusf(float x) {
  return (x > 20.f) ? x : log1pf(__expf(x));
}
__device__ __forceinline__ float sigmoidf(float x) {
  return 1.f / (1.f + __expf(-x));
}
__device__ __forceinline__ float safe_powf(float base, float p) {
  return (base > 0.f) ? __expf(p * __logf(base)) : 0.f;
}

// ---------------------------------------------------------------------------
// Kernel 1: y = x @ [w_qg | w_kv]  (M=BN, K=512, Nout=2048).
// Wave = 16x64 tile (4 accumulators sharing one A fragment).
// blockIdx.x = output region: 0->q, 1->g, 2->k, 3->v  (uniform epilogue).
// ---------------------------------------------------------------------------
__global__ __launch_bounds__(256)
void k1_qkvg(const float* __restrict__ x, const float* __restrict__ w_qg,
             const float* __restrict__ w_kv, const float* __restrict__ power,
             const float* __restrict__ scale_p, const float* __restrict__ pos_enc,
             float* __restrict__ q_pos, float* __restrict__ q_neg,
             float* __restrict__ k_pos, float* __restrict__ k_neg,
             float* __restrict__ vbuf, float* __restrict__ gbuf)
{
  __shared__ float xs[16 * 516];             // padded stride: conflict-free A reads
  const int tid = threadIdx.x;
  const int m0 = blockIdx.y * 16;
  const int region = blockIdx.x;             // 0:q 1:g 2:k 3:v

  for (int i = tid; i < 16 * 512; i += 256) {
    int r = i >> 9, k = i & 511;
    xs[r * 516 + k] = x[(size_t)(m0 + r) * C_ + k];
  }
  __syncthreads();

  const int wave = tid >> 5, lane = tid & 31;
  const int oc0 = region * 512 + wave * 64;  // first of this wave's 64 cols
  const float* W = (region < 2) ? w_qg : w_kv;
  const int wc0 = (region < 2) ? oc0 : (oc0 - 1024);

  const int ar    = lane & 15;               // A row (M)
  const int koff  = (lane >> 4) * 2;         // K sub-offset
  const int bcol0 = wc0 + (lane & 15);       // B column base

  v8f acc[4] = {};
  for (int kb = 0; kb < 512; kb += 4) {
    int k = kb + koff;
    v2f a; a.x = xs[ar * 516 + k]; a.y = xs[ar * 516 + k + 1];
#pragma unroll
    for (int t = 0; t < 4; ++t) {
      v2f bf;
      bf.x = W[(size_t)k * 1024 + bcol0 + t * 16];
      bf.y = W[(size_t)(k + 1) * 1024 + bcol0 + t * 16];
      WMMA_F32(acc[t], a, bf);
    }
  }

  const int lf = lane & 15;
  const int rb = (lane >> 4) * 8;
  if (region == 0) {                         // ---- q ----
#pragma unroll
    for (int t = 0; t < 4; ++t) {
      int c = wave * 64 + t * 16 + lf;
      float sc = softplusf(scale_p[c]);
      float pw = 1.f + ALPHA * sigmoidf(power[c]);
      for (int r = 0; r < 8; ++r) {
        size_t m = (size_t)(m0 + rb + r);
        float qv = acc[t][r] / sc;
        q_pos[m * C_ + c] = safe_powf(qv, pw);
        q_neg[m * C_ + c] = safe_powf(-qv, pw);
      }
    }
  } else if (region == 1) {                  // ---- g ----
#pragma unroll
    for (int t = 0; t < 4; ++t) {
      int c = wave * 64 + t * 16 + lf;
      for (int r = 0; r < 8; ++r) {
        size_t m = (size_t)(m0 + rb + r);
        gbuf[m * C_ + c] = acc[t][r];
      }
    }
  } else if (region == 2) {                  // ---- k ----
#pragma unroll
    for (int t = 0; t < 4; ++t) {
      int c = wave * 64 + t * 16 + lf;
      float sc = softplusf(scale_p[c]);
      float pw = 1.f + ALPHA * sigmoidf(power[c]);
      for (int r = 0; r < 8; ++r) {
        size_t m = (size_t)(m0 + rb + r);
        int n = (int)(m & (N_ - 1));
        float kv = (acc[t][r] + pos_enc[(size_t)n * C_ + c]) / sc;
        k_pos[m * C_ + c] = safe_powf(kv, pw);
        k_neg[m * C_ + c] = safe_powf(-kv, pw);
      }
    }
  } else {                                   // ---- v ----
#pragma unroll
    for (int t = 0; t < 4; ++t) {
      int c = wave * 64 + t * 16 + lf;
      for (int r = 0; r < 8; ++r) {
        size_t m = (size_t)(m0 + rb + r);
        vbuf[m * C_ + c] = acc[t][r];
      }
    }
  }
}

// ---------------------------------------------------------------------------
// Kernel 2: per (b,h): k_mean[128] and  B'[128][64] = k_cat^T @ v / N with the
// q_opp half-swap pre-applied (f<32 -> row e ; f>=32 -> row e^64).
// ---------------------------------------------------------------------------
__global__ __launch_bounds__(256)
void k2_kv(const float* __restrict__ k_pos, const float* __restrict__ k_neg,
           const float* __restrict__ vbuf,
           float* __restrict__ kmean, float* __restrict__ kvB)
{
  __shared__ float red[256];
  const int bh = blockIdx.x;
  const int b = bh >> 3, h = bh & 7;
  const int tid = threadIdx.x;
  const float invN = 1.f / (float)N_;

  { // phase 1: k_mean
    int e = tid & 127, half = tid >> 7;
    const float* src = ((e < 64) ? k_pos : k_neg)
        + ((size_t)b * N_ + (size_t)half * 2048) * C_ + h * 64 + (e & 63);
    float s = 0.f;
    for (int i = 0; i < 2048; ++i) s += src[(size_t)i * C_];
    red[tid] = s;
    __syncthreads();
    if (tid < 128) kmean[(size_t)bh * 128 + tid] = (red[tid] + red[tid + 128]) * invN;
    __syncthreads();
  }

  // phase 2: (e x n) @ (n x f) reduction GEMM, K = N = 4096
  const int wave = tid >> 5, lane = tid & 31;
  const float* kbase = (wave < 4) ? k_pos : k_neg;
  const int e_l = (wave * 16 + (lane & 15)) & 63;
  const int koff = (lane >> 4) * 2;
  const float* kptr = kbase + (size_t)b * N_ * C_ + h * 64 + e_l;
  const float* vptr = vbuf + (size_t)b * N_ * C_ + h * 64 + (lane & 15);

  v8f acc[4] = {};
  for (int nb = 0; nb < N_; nb += 4) {
    size_t n = (size_t)(nb + koff);
    v2f a; a.x = kptr[n * C_]; a.y = kptr[(n + 1) * C_];
#pragma unroll
    for (int t = 0; t < 4; ++t) {
      v2f bf;
      bf.x = vptr[n * C_ + t * 16];
      bf.y = vptr[(n + 1) * C_ + t * 16];
      WMMA_F32(acc[t], a, bf);
    }
  }

  float* out = kvB + (size_t)bh * 128 * 64;
  const int rb = (lane >> 4) * 8;
#pragma unroll
  for (int ft = 0; ft < 4; ++ft) {
    int f = ft * 16 + (lane & 15);
    for (int r = 0; r < 8; ++r) {
      int e = wave * 16 + rb + r;
      int row = (f < 32) ? e : (e ^ 64);    // pre-apply q_opp half swap
      out[row * 64 + f] = acc[ft][r] * invN;
    }
  }
}

// ---------------------------------------------------------------------------
// Kernel 3: 5x5 depthwise conv over v viewed as (b, c) 64x64 images.
// ---------------------------------------------------------------------------
__global__ __launch_bounds__(256)
void k3_dwc(const float* __restrict__ vbuf, const float* __restrict__ dwc_w,
            const float* __restrict__ dwc_b, float* __restrict__ vdwc)
{
  size_t gid = (size_t)blockIdx.x * 256 + threadIdx.x;   // over BN*C
  int c = (int)(gid & (C_ - 1));
  size_t m = gid >> 9;                                   // b*N + n
  int n = (int)(m & (N_ - 1));
  size_t b = m >> 12;
  int y = n >> 6, x0 = n & 63;
  int d = c & 63;
  float s = dwc_b[d];
  const float* wv = dwc_w + d * 25;
  const float* vb = vbuf + (b * (size_t)N_) * C_ + c;
  for (int dy = -2; dy <= 2; ++dy) {
    int yy = y + dy;
    if (yy < 0 || yy > 63) continue;
    for (int dx = -2; dx <= 2; ++dx) {
      int xw = x0 + dx;
      if (xw < 0 || xw > 63) continue;
      s += vb[(size_t)(yy * 64 + xw) * C_] * wv[(dy + 2) * 5 + (dx + 2)];
    }
  }
  vdwc[gid] = s;
}

// ---------------------------------------------------------------------------
// Kernel 4: per (b,h,32-token tile): x = q_sim @ B' (128 -> 64), z-normalize,
// fuse (xa + v_dwc) * g, write pre-projection activations.
// ---------------------------------------------------------------------------
__global__ __launch_bounds__(256)
void k4_attn(const float* __restrict__ q_pos, const float* __restrict__ q_neg,
             const float* __restrict__ kmean, const float* __restrict__ kvB,
             const float* __restrict__ vdwc, const float* __restrict__ gbuf,
             float* __restrict__ tmp)
{
  __shared__ float qs[32 * 132];   // padded: A-fragment reads conflict-free
  __shared__ float kvs[128 * 80];  // padded: B-fragment reads conflict-free
  __shared__ float km[128];
  __shared__ float zbuf[64];
  const int tid = threadIdx.x;
  const int bh = blockIdx.y;
  const int b = bh >> 3, h = bh & 7;
  const int n0 = blockIdx.x * 32;
  const size_t base = ((size_t)b * N_ + n0) * C_ + h * 64;

  for (int i = tid; i < 32 * 128; i += 256) {
    int tok = i >> 7, e = i & 127;
    qs[tok * 132 + e] = ((e < 64) ? q_pos : q_neg)[base + (size_t)tok * C_ + (e & 63)];
  }
  for (int i = tid; i < 128 * 64; i += 256) {
    int e = i >> 6, f = i & 63;
    kvs[e * 80 + f] = kvB[(size_t)bh * 8192 + i];
  }
  if (tid < 128) km[tid] = kmean[(size_t)bh * 128 + tid];
  __syncthreads();

  if (tid < 64) {                  // z_sim / z_opp per token
    int tok = tid >> 1, op = tid & 1;
    float s = 0.f;
    for (int e = 0; e < 128; ++e)
      s += qs[tok * 132 + e] * km[op ? (e ^ 64) : e];
    zbuf[tok * 2 + op] = 1.f / (s + EPS_);
  }
  __syncthreads();

  const int wave = tid >> 5, lane = tid & 31;
  const int sub = wave >> 2;       // token subtile (0/1)
  const int ft  = wave & 3;        // f tile (0..3)
  const int tokA = sub * 16 + (lane & 15);
  const int koff = (lane >> 4) * 2;
  const int fB   = ft * 16 + (lane & 15);

  v8f acc = {};
  for (int eb = 0; eb < 128; eb += 4) {
    int e = eb + koff;
    v2f a;  a.x  = qs[tokA * 132 + e]; a.y  = qs[tokA * 132 + e + 1];
    v2f bb; bb.x = kvs[e * 80 + fB];   bb.y = kvs[(e + 1) * 80 + fB];
    WMMA_F32(acc, a, bb);
  }

  const int rb = (lane >> 4) * 8;
  const int f = ft * 16 + (lane & 15);
  for (int r = 0; r < 8; ++r) {
    int tok = sub * 16 + rb + r;
    float z = zbuf[tok * 2 + (f < 32 ? 0 : 1)];
    size_t idx = base + (size_t)tok * C_ + f;
    tmp[idx] = (acc[r] * z + vdwc[idx]) * gbuf[idx];
  }
}

// ---------------------------------------------------------------------------
// Kernel 5: out = tmp @ w_proj + b_proj  (M=BN, K=512, N=512).
// Wave = 16x64 tile (4 accumulators sharing one A fragment).
// ---------------------------------------------------------------------------
__global__ __launch_bounds__(256)
void k5_proj(const float* __restrict__ tmp, const float* __restrict__ w_proj,
             const float* __restrict__ b_proj, float* __restrict__ out)
{
  __shared__ float xs[16 * 516];
  const int tid = threadIdx.x;
  const int m0 = blockIdx.y * 16;
  for (int i = tid; i < 16 * 512; i += 256) {
    int r = i >> 9, k = i & 511;
    xs[r * 516 + k] = tmp[(size_t)(m0 + r) * C_ + k];
  }
  __syncthreads();

  const int wave = tid >> 5, lane = tid & 31;
  const int oc0 = wave * 64;                 // this wave's 64 output cols
  const int ar    = lane & 15;
  const int koff  = (lane >> 4) * 2;
  const int bcol0 = oc0 + (lane & 15);

  v8f acc[4] = {};
  for (int kb = 0; kb < 512; kb += 4) {
    int k = kb + koff;
    v2f a; a.x = xs[ar * 516 + k]; a.y = xs[ar * 516 + k + 1];
#pragma unroll
    for (int t = 0; t < 4; ++t) {
      v2f bf;
      bf.x = w_proj[(size_t)k * C_ + bcol0 + t * 16];
      bf.y = w_proj[(size_t)(k + 1) * C_ + bcol0 + t * 16];
      WMMA_F32(acc[t], a, bf);
    }
  }

  const int rb = (lane >> 4) * 8;
#pragma unroll
  for (int t = 0; t < 4; ++t) {
    int c = oc0 + t * 16 + (lane & 15);
    float bias = b_proj[c];
    for (int r = 0; r < 8; ++r) {
      size_t m = (size_t)(m0 + rb + r);
      out[m * C_ + c] = acc[t][r] + bias;
    }
  }
}

// ---------------------------------------------------------------------------
extern "C" void kernel_launch(void* const* d_in, const int* in_sizes, int n_in,
                              void* d_out, int out_size, void* d_ws, size_t ws_size,
                              hipStream_t stream) {
  (void)in_sizes; (void)n_in; (void)out_size; (void)ws_size;
  const float* x       = (const float*)d_in[0];
  const float* w_qg    = (const float*)d_in[1];
  const float* w_kv    = (const float*)d_in[2];
  const float* w_proj  = (const float*)d_in[3];
  const float* b_proj  = (const float*)d_in[4];
  const float* dwc_w   = (const float*)d_in[5];
  const float* dwc_b   = (const float*)d_in[6];
  const float* power   = (const float*)d_in[7];
  const float* scale_p = (const float*)d_in[8];
  const float* pos_enc = (const float*)d_in[9];
  // d_in[10]/d_in[11] = H, W = 64 (hardcoded)

  float* ws = (float*)d_ws;
  const size_t S = (size_t)BN * C_;   // 16,777,216 floats per buffer
  float* q_pos = ws;
  float* q_neg = q_pos + S;
  float* k_pos = q_neg + S;
  float* k_neg = k_pos + S;
  float* vbuf  = k_neg + S;
  float* gbuf  = vbuf + S;
  float* vdwc  = gbuf + S;
  float* tmp   = vdwc + S;
  float* kmean = tmp + S;                 // 64 * 128
  float* kvB   = kmean + 64 * 128;        // 64 * 128 * 64

  k1_qkvg<<<dim3(4, BN / 16), 256, 0, stream>>>(x, w_qg, w_kv, power, scale_p, pos_enc,
                                                q_pos, q_neg, k_pos, k_neg, vbuf, gbuf);
  k2_kv<<<dim3(64), 256, 0, stream>>>(k_pos, k_neg, vbuf, kmean, kvB);
  k3_dwc<<<dim3((unsigned)(S / 256)), 256, 0, stream>>>(vbuf, dwc_w, dwc_b, vdwc);
  k4_attn<<<dim3(N_ / 32, 64), 256, 0, stream>>>(q_pos, q_neg, kmean, kvB, vdwc, gbuf, tmp);
  k5_proj<<<dim3(1, BN / 16), 256, 0, stream>>>(tmp, w_proj, b_proj, (float*)d_out);
}